// CellLineGraphEncoder_50087908606360
// MI455X (gfx1250) — compile-verified
//
#include <hip/hip_runtime.h>
#include <hip/hip_bf16.h>

// ---------------------------------------------------------------------------
// CDNA5 (gfx1250) GENConv graph encoder.
//  - fp32 WMMA (v_wmma_f32_16x16x4_f32) for all node-side GEMMs
//  - rank-1 edge-feature recomputation (never materialize [E,64])
//  - 2-pass scatter-softmax with ordered-uint atomic-max keys
//  - wave-per-edge layout with scalar (readfirstlane) index loads
// ---------------------------------------------------------------------------

typedef __attribute__((ext_vector_type(2))) float v2f;
typedef __attribute__((ext_vector_type(8))) float v8f;

#define HID 64
#define NEG_INF_KEY 0x007FFFFFu   // ordKey(-inf)

__device__ __forceinline__ unsigned ordKey(float f) {
    unsigned u = __float_as_uint(f);
    return u ^ ((unsigned)((int)u >> 31) | 0x80000000u);
}
__device__ __forceinline__ float keyOrd(unsigned k) {
    unsigned u = (k & 0x80000000u) ? (k ^ 0x80000000u) : ~k;
    return __uint_as_float(u);
}

__device__ __forceinline__ v8f wmma4(v2f a, v2f b, v8f c) {
    // D(16x16,f32) += A(16x4,f32) * B(4x16,f32)
    return __builtin_amdgcn_wmma_f32_16x16x4_f32(
        false, a, false, b, (short)0, c, false, false);
}

// ---------------------------------------------------------------------------
// h0 = x[N,16] @ W_node[16,64] + b_node   (one 16-node tile per block, 4 waves)
// ---------------------------------------------------------------------------
__global__ void k_embed(const float* __restrict__ x, const float* __restrict__ Wn,
                        const float* __restrict__ bn, float* __restrict__ z, int N) {
    __shared__ float xt[16 * 16];
    const int tid = threadIdx.x, tile = blockIdx.x;
    const bool fullTile = (tile * 16 + 16) <= N;
    for (int idx = tid; idx < 256; idx += blockDim.x) {
        int node = tile * 16 + (idx >> 4);
        xt[idx] = (node < N) ? x[(size_t)node * 16 + (idx & 15)] : 0.f;
    }
    __syncthreads();
    const int wave = tid >> 5, lane = tid & 31;
    const int row = lane & 15, kh = lane >> 4, col = lane & 15;
    const int n0 = wave * 16;
    v8f acc = {};
    for (int kk = 0; kk < 16; kk += 4) {
        int k = kk + kh * 2;
        v2f a = { xt[row * 16 + k], xt[row * 16 + k + 1] };
        v2f b = { Wn[k * HID + n0 + col], Wn[(k + 1) * HID + n0 + col] };
        acc = wmma4(a, b, acc);
    }
    const float bias = bn[n0 + col];
    float* zp = z + ((size_t)(tile * 16 + kh * 8)) * HID + n0 + col;
    if (fullTile) {
        #pragma unroll
        for (int i = 0; i < 8; ++i) zp[(size_t)i * HID] = acc[i] + bias;
    } else {
        for (int i = 0; i < 8; ++i) {
            int node = tile * 16 + kh * 8 + i;
            if (node < N) z[(size_t)node * HID + n0 + col] = acc[i] + bias;
        }
    }
}

// ---------------------------------------------------------------------------
// z = relu(layernorm(h; g, b))   one node per thread
// ---------------------------------------------------------------------------
__global__ void k_ln_relu(const float* __restrict__ h, const float* __restrict__ g,
                          const float* __restrict__ b, float* __restrict__ z, int N) {
    int node = blockIdx.x * blockDim.x + threadIdx.x;
    const int stride = gridDim.x * blockDim.x;
    for (; node < N; node += stride) {
        const float4* hr = reinterpret_cast<const float4*>(h + (size_t)node * HID);
        float4 v[16];
        float s = 0.f, sq = 0.f;
        #pragma unroll
        for (int i = 0; i < 16; ++i) {
            v[i] = hr[i];
            s  += v[i].x + v[i].y + v[i].z + v[i].w;
            sq += v[i].x * v[i].x + v[i].y * v[i].y + v[i].z * v[i].z + v[i].w * v[i].w;
        }
        float mu = s * (1.f / HID);
        float rs = rsqrtf(sq * (1.f / HID) - mu * mu + 1e-5f);
        float* zr = z + (size_t)node * HID;
        #pragma unroll
        for (int i = 0; i < 16; ++i) {
            int c = i * 4;
            zr[c + 0] = fmaxf((v[i].x - mu) * rs * g[c + 0] + b[c + 0], 0.f);
            zr[c + 1] = fmaxf((v[i].y - mu) * rs * g[c + 1] + b[c + 1], 0.f);
            zr[c + 2] = fmaxf((v[i].z - mu) * rs * g[c + 2] + b[c + 2], 0.f);
            zr[c + 3] = fmaxf((v[i].w - mu) * rs * g[c + 3] + b[c + 3], 0.f);
        }
    }
}

// ---------------------------------------------------------------------------
// per-layer accumulator reset
// ---------------------------------------------------------------------------
__global__ void k_init(unsigned* __restrict__ mkey, float* __restrict__ s,
                       float* __restrict__ w, int n) {
    int i = blockIdx.x * blockDim.x + threadIdx.x;
    if (i < n) { mkey[i] = NEG_INF_KEY; s[i] = 0.f; w[i] = 0.f; }
}

// ---------------------------------------------------------------------------
// Edge pass 1: segment-max of logits. One wave per edge, 2 channels per lane.
// Edge index forced wave-uniform -> scalar loads of src/dst/edge_attr.
// ---------------------------------------------------------------------------
__global__ void k_edge_max(const int* __restrict__ src, const int* __restrict__ dst,
                           const float* __restrict__ ea, const float* __restrict__ We,
                           const float* __restrict__ be, const float* __restrict__ z,
                           const float* __restrict__ tp, unsigned* __restrict__ mkey,
                           int E) {
    int gid = blockIdx.x * blockDim.x + threadIdx.x;
    int e = __builtin_amdgcn_readfirstlane(gid >> 5);   // wave-uniform edge id
    if (e >= E) return;
    int c = (gid & 31) * 2;
    if (e + 4096 < E) {
        __builtin_prefetch(&ea[e + 4096], 0, 1);
        __builtin_prefetch(&src[e + 4096], 0, 1);
        __builtin_prefetch(&dst[e + 4096], 0, 1);
    }
    const float t = tp[0];
    const float a = ea[e];
    const int si = src[e], di = dst[e];
    float2 zv = *reinterpret_cast<const float2*>(z + (size_t)si * HID + c);
    float msg0 = fmaxf(zv.x + (a * We[c]     + be[c]),     0.f) + 1e-7f;
    float msg1 = fmaxf(zv.y + (a * We[c + 1] + be[c + 1]), 0.f) + 1e-7f;
    unsigned* mrow = mkey + (size_t)di * HID + c;
    atomicMax(&mrow[0], ordKey(msg0 * t));
    atomicMax(&mrow[1], ordKey(msg1 * t));
}

// ---------------------------------------------------------------------------
// Edge pass 2: e = exp(logit - m); accumulate  s += e  and  wsum += e*msg.
// ---------------------------------------------------------------------------
__global__ void k_edge_sum(const int* __restrict__ src, const int* __restrict__ dst,
                           const float* __restrict__ ea, const float* __restrict__ We,
                           const float* __restrict__ be, const float* __restrict__ z,
                           const float* __restrict__ tp,
                           const unsigned* __restrict__ mkey,
                           float* __restrict__ sden, float* __restrict__ wsum, int E) {
    int gid = blockIdx.x * blockDim.x + threadIdx.x;
    int e = __builtin_amdgcn_readfirstlane(gid >> 5);   // wave-uniform edge id
    if (e >= E) return;
    int c = (gid & 31) * 2;
    if (e + 4096 < E) {
        __builtin_prefetch(&ea[e + 4096], 0, 1);
        __builtin_prefetch(&src[e + 4096], 0, 1);
        __builtin_prefetch(&dst[e + 4096], 0, 1);
    }
    const float t = tp[0];
    const float a = ea[e];
    const int si = src[e], di = dst[e];
    float2 zv = *reinterpret_cast<const float2*>(z + (size_t)si * HID + c);
    float msg0 = fmaxf(zv.x + (a * We[c]     + be[c]),     0.f) + 1e-7f;
    float msg1 = fmaxf(zv.y + (a * We[c + 1] + be[c + 1]), 0.f) + 1e-7f;
    size_t o = (size_t)di * HID + c;
    unsigned k0 = mkey[o], k1 = mkey[o + 1];
    float mx0 = (k0 == NEG_INF_KEY) ? 0.f : keyOrd(k0);
    float mx1 = (k1 == NEG_INF_KEY) ? 0.f : keyOrd(k1);
    float e0 = __expf(msg0 * t - mx0);
    float e1 = __expf(msg1 * t - mx1);
    atomicAdd(&sden[o],     e0);
    atomicAdd(&sden[o + 1], e1);
    atomicAdd(&wsum[o],     e0 * msg0);
    atomicAdd(&wsum[o + 1], e1 * msg1);
}

// ---------------------------------------------------------------------------
// Node phase, fused WMMA MLP per 16-node tile (256 threads = 8 waves):
//   hin = wsum/(s+eps) + z
//   t1  = relu(LN(hin @ W1 + b1))
//   out = t1 @ W2 + b2 ;  h = (res ? h : 0) + out
// ---------------------------------------------------------------------------
__global__ void k_finalize(const float* __restrict__ wsum, const float* __restrict__ sden,
                           const float* __restrict__ z, float* __restrict__ h,
                           const float* __restrict__ W1, const float* __restrict__ b1,
                           const float* __restrict__ g1, const float* __restrict__ be1,
                           const float* __restrict__ W2, const float* __restrict__ b2,
                           int addres, int N) {
    __shared__ float hin[16 * HID];
    __shared__ float t1[16 * 2 * HID];
    __shared__ float mu[16], rstd[16];
    const int tid = threadIdx.x, tile = blockIdx.x;
    const bool fullTile = (tile * 16 + 16) <= N;

    if (fullTile) {
        const size_t base = (size_t)(tile * 16) * HID;
        for (int idx = tid; idx < 16 * HID; idx += blockDim.x)
            hin[idx] = wsum[base + idx] / (sden[base + idx] + 1e-16f) + z[base + idx];
    } else {
        for (int idx = tid; idx < 16 * HID; idx += blockDim.x) {
            int row = idx >> 6, c = idx & 63, node = tile * 16 + row;
            float v = 0.f;
            if (node < N) {
                size_t o = (size_t)node * HID + c;
                v = wsum[o] / (sden[o] + 1e-16f) + z[o];
            }
            hin[idx] = v;
        }
    }
    __syncthreads();

    const int wave = tid >> 5, lane = tid & 31;
    const int row = lane & 15, kh = lane >> 4, col = lane & 15;

    { // GEMM1: [16x64] @ [64x128] -> t1, 8 waves x one 16x16 tile
        const int n0 = wave * 16;
        v8f acc = {};
        for (int kk = 0; kk < HID; kk += 4) {
            int k = kk + kh * 2;
            v2f a = { hin[row * HID + k], hin[row * HID + k + 1] };
            v2f b = { W1[(size_t)k * 128 + n0 + col],
                      W1[(size_t)(k + 1) * 128 + n0 + col] };
            acc = wmma4(a, b, acc);
        }
        #pragma unroll
        for (int i = 0; i < 8; ++i) {
            int r = i + kh * 8;
            t1[r * 128 + n0 + col] = acc[i] + b1[n0 + col];
        }
    }
    __syncthreads();

    if (tid < 16) { // LN stats over 128-wide hidden
        float s = 0.f, sq = 0.f;
        for (int c = 0; c < 128; ++c) { float v = t1[tid * 128 + c]; s += v; sq += v * v; }
        float m = s * (1.f / 128.f);
        mu[tid] = m;
        rstd[tid] = rsqrtf(sq * (1.f / 128.f) - m * m + 1e-5f);
    }
    __syncthreads();

    for (int idx = tid; idx < 16 * 128; idx += blockDim.x) {
        int r = idx >> 7, c = idx & 127;
        float v = (t1[idx] - mu[r]) * rstd[r] * g1[c] + be1[c];
        t1[idx] = fmaxf(v, 0.f);
    }
    __syncthreads();

    if (wave < 4) { // GEMM2: [16x128] @ [128x64] -> h
        const int n0 = wave * 16;
        v8f acc = {};
        for (int kk = 0; kk < 128; kk += 4) {
            int k = kk + kh * 2;
            v2f a = { t1[row * 128 + k], t1[row * 128 + k + 1] };
            v2f b = { W2[(size_t)k * HID + n0 + col],
                      W2[(size_t)(k + 1) * HID + n0 + col] };
            acc = wmma4(a, b, acc);
        }
        const float bias = b2[n0 + col];
        float* hp = h + ((size_t)(tile * 16 + kh * 8)) * HID + n0 + col;
        if (fullTile) {
            if (addres) {
                #pragma unroll
                for (int i = 0; i < 8; ++i)
                    hp[(size_t)i * HID] += acc[i] + bias;
            } else {
                #pragma unroll
                for (int i = 0; i < 8; ++i)
                    hp[(size_t)i * HID] = acc[i] + bias;
            }
        } else {
            for (int i = 0; i < 8; ++i) {
                int node = tile * 16 + kh * 8 + i;
                if (node < N) {
                    size_t o = (size_t)node * HID + n0 + col;
                    float v = acc[i] + bias;
                    h[o] = addres ? (h[o] + v) : v;
                }
            }
        }
    }
}

// ---------------------------------------------------------------------------
// Global pooling of relu(LN(h; norm_g[0], norm_b[0]))
// ---------------------------------------------------------------------------
__global__ void k_pool_init(float* psum, unsigned* pmax) {
    int i = threadIdx.x;
    if (i < HID) { psum[i] = 0.f; pmax[i] = NEG_INF_KEY; }
}

__global__ void k_pool(const float* __restrict__ h, const float* __restrict__ g,
                       const float* __restrict__ b, float* __restrict__ psum,
                       unsigned* __restrict__ pmax, int N) {
    __shared__ float ssum[HID];
    __shared__ unsigned smax[HID];
    const int tid = threadIdx.x;
    if (tid < HID) { ssum[tid] = 0.f; smax[tid] = NEG_INF_KEY; }
    __syncthreads();
    for (int node = blockIdx.x * blockDim.x + tid; node < N;
         node += gridDim.x * blockDim.x) {
        const float4* hr = reinterpret_cast<const float4*>(h + (size_t)node * HID);
        float4 v[16];
        float s = 0.f, sq = 0.f;
        #pragma unroll
        for (int i = 0; i < 16; ++i) {
            v[i] = hr[i];
            s  += v[i].x + v[i].y + v[i].z + v[i].w;
            sq += v[i].x * v[i].x + v[i].y * v[i].y + v[i].z * v[i].z + v[i].w * v[i].w;
        }
        float mval = s * (1.f / HID);
        float rs = rsqrtf(sq * (1.f / HID) - mval * mval + 1e-5f);
        #pragma unroll
        for (int i = 0; i < 16; ++i) {
            int c = i * 4;
            float o0 = fmaxf((v[i].x - mval) * rs * g[c + 0] + b[c + 0], 0.f);
            float o1 = fmaxf((v[i].y - mval) * rs * g[c + 1] + b[c + 1], 0.f);
            float o2 = fmaxf((v[i].z - mval) * rs * g[c + 2] + b[c + 2], 0.f);
            float o3 = fmaxf((v[i].w - mval) * rs * g[c + 3] + b[c + 3], 0.f);
            atomicAdd(&ssum[c + 0], o0); atomicMax(&smax[c + 0], ordKey(o0));
            atomicAdd(&ssum[c + 1], o1); atomicMax(&smax[c + 1], ordKey(o1));
            atomicAdd(&ssum[c + 2], o2); atomicMax(&smax[c + 2], ordKey(o2));
            atomicAdd(&ssum[c + 3], o3); atomicMax(&smax[c + 3], ordKey(o3));
        }
    }
    __syncthreads();
    if (tid < HID) {
        atomicAdd(&psum[tid], ssum[tid]);
        atomicMax(&pmax[tid], smax[tid]);
    }
}

// ---------------------------------------------------------------------------
// Head: channel-bin avg/max pooling -> emb[64] @ W_lin[64,64] + b_lin
// ---------------------------------------------------------------------------
__global__ void k_head(const float* __restrict__ psum, const unsigned* __restrict__ pmax,
                       const float* __restrict__ Wl, const float* __restrict__ bl,
                       float* __restrict__ out, int N) {
    __shared__ float emb[64];
    const int tid = threadIdx.x; // 64 threads
    if (tid < 32) {
        float a0 = psum[2 * tid] / (float)N, a1 = psum[2 * tid + 1] / (float)N;
        emb[tid] = 0.5f * (a0 + a1);
        float x0 = keyOrd(pmax[2 * tid]), x1 = keyOrd(pmax[2 * tid + 1]);
        emb[32 + tid] = fmaxf(x0, x1);
    }
    __syncthreads();
    float acc = bl[tid];
    for (int k = 0; k < 64; ++k) acc += emb[k] * Wl[k * 64 + tid];
    out[tid] = acc;
}

// ---------------------------------------------------------------------------
extern "C" void kernel_launch(void* const* d_in, const int* in_sizes, int n_in,
                              void* d_out, int out_size, void* d_ws, size_t ws_size,
                              hipStream_t stream) {
    const float* x         = (const float*)d_in[0];
    const float* edge_attr = (const float*)d_in[1];
    const int*   ei        = (const int*)  d_in[2];
    const float* W_node    = (const float*)d_in[3];
    const float* b_node    = (const float*)d_in[4];
    const float* W_edge    = (const float*)d_in[5];
    const float* b_edge    = (const float*)d_in[6];
    const float* t         = (const float*)d_in[7];
    const float* W1        = (const float*)d_in[8];
    const float* b1        = (const float*)d_in[9];
    const float* ln_g      = (const float*)d_in[10];
    const float* ln_b      = (const float*)d_in[11];
    const float* W2        = (const float*)d_in[12];
    const float* b2        = (const float*)d_in[13];
    const float* norm_g    = (const float*)d_in[14];
    const float* norm_b    = (const float*)d_in[15];
    const float* W_lin     = (const float*)d_in[16];
    const float* b_lin     = (const float*)d_in[17];

    const int N = in_sizes[0] / 16;
    const int E = in_sizes[1];
    const int* src = ei;
    const int* dst = ei + E;

    // workspace layout (all L2-resident: ~64 MB total)
    const size_t nfeat = (size_t)N * HID;
    float*    z     = (float*)d_ws;
    float*    h     = z + nfeat;
    float*    sden  = h + nfeat;
    float*    wsumb = sden + nfeat;
    unsigned* mkey  = (unsigned*)(wsumb + nfeat);
    float*    psum  = (float*)(mkey + nfeat);
    unsigned* pmax  = (unsigned*)(psum + HID);
    (void)ws_size; (void)n_in; (void)out_size;

    const int ntiles     = (N + 15) / 16;
    const int initBlocks = (int)((nfeat + 255) / 256);
    const int edgeBlocks = (E * 32 + 255) / 256;
    const int nodeBlocks = (N + 255) / 256;

    k_embed<<<ntiles, 128, 0, stream>>>(x, W_node, b_node, z, N);

    for (int l = 0; l < 3; ++l) {
        if (l > 0)
            k_ln_relu<<<nodeBlocks, 256, 0, stream>>>(
                h, norm_g + (size_t)l * HID, norm_b + (size_t)l * HID, z, N);
        k_init<<<initBlocks, 256, 0, stream>>>(mkey, sden, wsumb, (int)nfeat);
        k_edge_max<<<edgeBlocks, 256, 0, stream>>>(
            src, dst, edge_attr, W_edge, b_edge, z, t + l, mkey, E);
        k_edge_sum<<<edgeBlocks, 256, 0, stream>>>(
            src, dst, edge_attr, W_edge, b_edge, z, t + l, mkey, sden, wsumb, E);
        k_finalize<<<ntiles, 256, 0, stream>>>(
            wsumb, sden, z, h,
            W1 + (size_t)l * HID * 2 * HID, b1 + (size_t)l * 2 * HID,
            ln_g + (size_t)l * 2 * HID, ln_b + (size_t)l * 2 * HID,
            W2 + (size_t)l * 2 * HID * HID, b2 + (size_t)l * HID,
            (l > 0) ? 1 : 0, N);
    }

    k_pool_init<<<1, 64, 0, stream>>>(psum, pmax);
    k_pool<<<128, 256, 0, stream>>>(h, norm_g, norm_b, psum, pmax, N);
    k_head<<<1, 64, 0, stream>>>(psum, pmax, W_lin, b_lin, (float*)d_out, N);
}